// SeqFirstSelfAttention_grad_7679401525960
// MI455X (gfx1250) — compile-verified
//
#include <hip/hip_runtime.h>
#include <hip/hip_bf16.h>
#include <stdint.h>

typedef __attribute__((ext_vector_type(2))) float v2f;
typedef __attribute__((ext_vector_type(8))) float v8f;
typedef int v4i_vs __attribute__((vector_size(4 * sizeof(int))));

#define S_LEN 2048
#define BATCH 2
#define HID   1024
#define NHEAD 16
#define DHEAD 64
#define NM    (BATCH * NHEAD)   // 32 merged batch*head slices
#define H3    (3 * HID)         // 3072

static __device__ __forceinline__ v8f wmma4(v2f a, v2f b, v8f c) {
  // D = A(16x4 f32) * B(4x16 f32) + C(16x16 f32)
  return __builtin_amdgcn_wmma_f32_16x16x4_f32(false, a, false, b, (short)0, c,
                                               false, false);
}

// Per-lane async copy of 16 bytes global -> LDS (GLOBAL_LOAD_ASYNC_TO_LDS_B128,
// tracked by ASYNCcnt). No VGPR staging, frees the vector pipe for WMMA.
static __device__ __forceinline__ void async_cp16(const float* g, float* l) {
#if __has_builtin(__builtin_amdgcn_global_load_async_to_lds_b128)
  __builtin_amdgcn_global_load_async_to_lds_b128(
      (__attribute__((address_space(1))) v4i_vs*)g,
      (__attribute__((address_space(3))) v4i_vs*)l, 0, 0);
#else
  // VDST = LDS byte address (generic LDS pointer low 32 bits), VADDR = 64-bit
  // global address, GV mode.
  uint32_t lds = (uint32_t)(uintptr_t)l;
  uint64_t ga  = (uint64_t)(uintptr_t)g;
  asm volatile("global_load_async_to_lds_b128 %0, %1, off"
               :: "v"(lds), "v"(ga) : "memory");
#endif
}

static __device__ __forceinline__ void wait_async0() {
#if __has_builtin(__builtin_amdgcn_s_wait_asynccnt)
  __builtin_amdgcn_s_wait_asynccnt(0);
#else
  asm volatile("s_wait_asynccnt 0x0" ::: "memory");
#endif
}

// ---------------------------------------------------------------------------
// Kernel 1: QKV projection  C[r, j] = X[r, :] . W[j, :] + bias[j]
//   X: [4096, 1024]  (hidden_states, row r = s*B + b)
//   W: [3072, 1024]  (W_qkv, row-major, K contiguous)  -> NT GEMM
// Results scattered into separated Q/K/V tensors laid out [m][s][d],
// with m = b*16 + h, so the attention kernel reads coalesced.
// ---------------------------------------------------------------------------
#define K1_LDA 36  // padded LDS row stride (mult of 4 for b128, conflict-free)

__global__ __launch_bounds__(256) void qkv_gemm_kernel(
    const float* __restrict__ X, const float* __restrict__ W,
    const float* __restrict__ bias, float* __restrict__ Qb,
    float* __restrict__ Kb, float* __restrict__ Vb) {
  __shared__ float As[64 * K1_LDA];
  __shared__ float Bs[128 * K1_LDA];

  const int tid  = threadIdx.x;
  const int lane = tid & 31;
  const int w    = tid >> 5;  // 0..7
  const int wm   = w & 3;     // M sub-tile (16 rows each -> 64)
  const int wn   = w >> 2;    // N half (64 cols each -> 128)
  const int lo   = lane & 15;
  const int hi   = lane >> 4;

  const int m_base = blockIdx.y * 64;
  const int n_base = blockIdx.x * 128;

  v8f acc[4] = {};

  const int arow = tid >> 2, aq = (tid & 3) * 8;
  const int brow = tid >> 1, bq = (tid & 1) * 16;

  for (int kk = 0; kk < HID; kk += 32) {
    // A tile 64x32, B tile 128x32: async copy straight into LDS
    const float* ga = X + (size_t)(m_base + arow) * HID + kk + aq;
    float* la = &As[arow * K1_LDA + aq];
    async_cp16(ga, la);
    async_cp16(ga + 4, la + 4);
    const float* gb = W + (size_t)(n_base + brow) * HID + kk + bq;
    float* lb = &Bs[brow * K1_LDA + bq];
#pragma unroll
    for (int q = 0; q < 4; ++q) async_cp16(gb + q * 4, lb + q * 4);
    wait_async0();
    __syncthreads();

#pragma unroll
    for (int ks = 0; ks < 8; ++ks) {
      const int kf = ks * 4 + hi * 2;  // K position for this lane-half
      v2f a = *(const v2f*)&As[(wm * 16 + lo) * K1_LDA + kf];
#pragma unroll
      for (int nt = 0; nt < 4; ++nt) {
        v2f b = *(const v2f*)&Bs[(wn * 64 + nt * 16 + lo) * K1_LDA + kf];
        acc[nt] = wmma4(a, b, acc[nt]);
      }
    }
    __syncthreads();
  }

  // Epilogue: bias add + scatter into Q/K/V [m][s][d]
#pragma unroll
  for (int nt = 0; nt < 4; ++nt) {
    const int j   = n_base + wn * 64 + nt * 16 + lo;
    const float bj = bias[j];
    const int h   = j / 192;        // head (uniform across the 16-lane group)
    const int t3  = j % 192;
    const int seg = t3 >> 6;        // 0=Q 1=K 2=V
    const int d   = t3 & 63;
    float* dst = (seg == 0) ? Qb : (seg == 1) ? Kb : Vb;
#pragma unroll
    for (int i = 0; i < 8; ++i) {
      const int r = m_base + wm * 16 + i + 8 * hi;  // row in [S*B]
      const int s = r >> 1, b = r & 1;              // B == 2
      const int m = b * 16 + h;
      dst[((size_t)m * S_LEN + s) * DHEAD + d] = acc[nt][i] + bj;
    }
  }
}

// ---------------------------------------------------------------------------
// Kernel 2: per-row squared L2 norms of Q and K
// ---------------------------------------------------------------------------
__global__ __launch_bounds__(256) void row_norm_kernel(
    const float* __restrict__ Qb, const float* __restrict__ Kb,
    float* __restrict__ sq, float* __restrict__ sk) {
  const int idx = blockIdx.x * 256 + threadIdx.x;  // 0 .. 2*NM*S-1
  const int total = NM * S_LEN;                    // 65536
  const float* src = (idx < total) ? Qb : Kb;
  float* dst       = (idx < total) ? sq : sk;
  const int row = idx & (total - 1);
  const float* p = src + (size_t)row * DHEAD;
  float sum = 0.f;
#pragma unroll
  for (int i = 0; i < 16; ++i) {
    float4 v = *(const float4*)(p + i * 4);
    sum += v.x * v.x + v.y * v.y + v.z * v.z + v.w * v.w;
  }
  dst[row] = sum;
}

// ---------------------------------------------------------------------------
// Kernel 3: attention.  One block = one (slice m, 64-query block).
// 4 waves; wave w owns score/ctx rows [16w, 16w+16).
// K-tile LDS buffer is reused as the P (exp scores) tile.
// ---------------------------------------------------------------------------
#define K3_LD 68  // padded stride: mult of 4, conflict-free b64 frag reads

__global__ __launch_bounds__(128) void attention_kernel(
    const float* __restrict__ Qb, const float* __restrict__ Kb,
    const float* __restrict__ Vb, const float* __restrict__ sq,
    const float* __restrict__ sk, float* __restrict__ ctx_out,
    float* __restrict__ score_out) {
  __shared__ float Qs[64 * K3_LD];
  __shared__ float KPs[64 * K3_LD];  // K tile, reused as P tile
  __shared__ float Vs[64 * K3_LD];
  __shared__ float sqs[64];
  __shared__ float sks[64];

  const int m  = blockIdx.y;
  const int sb = blockIdx.x * 64;
  const int tid  = threadIdx.x;
  const int lane = tid & 31;
  const int w    = tid >> 5;  // 0..3
  const int lo   = lane & 15;
  const int hi   = lane >> 4;

  const int lrow = tid >> 1;        // 0..63
  const int lcol = (tid & 1) * 32;  // 0 or 32

  // Load Q block [64 x 64] (async) and its norms
  {
    const float* gq = Qb + ((size_t)m * S_LEN + sb + lrow) * DHEAD + lcol;
    float* lq = &Qs[lrow * K3_LD + lcol];
#pragma unroll
    for (int q = 0; q < 8; ++q) async_cp16(gq + q * 4, lq + q * 4);
    if (tid < 64) sqs[tid] = sq[(size_t)m * S_LEN + sb + tid];
  }

  v8f cacc[4] = {};
  float sqr[8];

  for (int t0 = 0; t0 < S_LEN; t0 += 64) {
    __syncthreads();  // previous iteration done with K/V/P tiles
    const float* gk = Kb + ((size_t)m * S_LEN + t0 + lrow) * DHEAD + lcol;
    const float* gv = Vb + ((size_t)m * S_LEN + t0 + lrow) * DHEAD + lcol;
    float* lk = &KPs[lrow * K3_LD + lcol];
    float* lv = &Vs[lrow * K3_LD + lcol];
#pragma unroll
    for (int q = 0; q < 8; ++q) {
      async_cp16(gk + q * 4, lk + q * 4);
      async_cp16(gv + q * 4, lv + q * 4);
    }
    if (tid < 64) sks[tid] = sk[(size_t)m * S_LEN + t0 + tid];
    wait_async0();     // covers Q block on iteration 0 as well (same counter)
    __syncthreads();   // publish K/V tiles (and Q) to all waves

    // ---- GEMM1: raw = Q . K^T  (M=s 16, N=t 64, K=d 64) ----
    v8f sacc[4] = {};
#pragma unroll
    for (int ks = 0; ks < 16; ++ks) {
      const int kf = ks * 4 + hi * 2;
      v2f a = *(const v2f*)&Qs[(w * 16 + lo) * K3_LD + kf];
#pragma unroll
      for (int nt = 0; nt < 4; ++nt) {
        v2f b = *(const v2f*)&KPs[(nt * 16 + lo) * K3_LD + kf];
        sacc[nt] = wmma4(a, b, sacc[nt]);
      }
    }

#pragma unroll
    for (int i = 0; i < 8; ++i) sqr[i] = sqs[w * 16 + i + 8 * hi];

    __syncthreads();  // everyone done reading K tile -> reuse as P tile

    // ---- epilogue: scores out (pre-exp) + P = exp(scores) into LDS ----
#pragma unroll
    for (int nt = 0; nt < 4; ++nt) {
      const int tcol  = nt * 16 + lo;
      const float skv = sks[tcol];
#pragma unroll
      for (int i = 0; i < 8; ++i) {
        const int srow = w * 16 + i + 8 * hi;
        const float sc =
            (sqr[i] + skv - 2.0f * sacc[nt][i]) * -0.0625f;  // -0.5/sqrt(64)
        score_out[((size_t)m * S_LEN + (sb + srow)) * S_LEN + t0 + tcol] = sc;
        KPs[srow * K3_LD + tcol] = __expf(sc);
      }
    }
    __syncthreads();  // P tile ready for all waves

    // ---- GEMM2: ctx += P . V  (M=s 16, N=d 64, K=t 64) ----
#pragma unroll
    for (int ks = 0; ks < 16; ++ks) {
      const int kf = ks * 4 + hi * 2;
      v2f a = *(const v2f*)&KPs[(w * 16 + lo) * K3_LD + kf];
#pragma unroll
      for (int dt = 0; dt < 4; ++dt) {
        v2f b;
        b.x = Vs[kf * K3_LD + dt * 16 + lo];
        b.y = Vs[(kf + 1) * K3_LD + dt * 16 + lo];
        cacc[dt] = wmma4(a, b, cacc[dt]);
      }
    }
  }

  // ---- write ctx: [S, B, H] with m = b*16 + h ----
  const int bb = m >> 4;
  const int hh = m & 15;
#pragma unroll
  for (int dt = 0; dt < 4; ++dt) {
#pragma unroll
    for (int i = 0; i < 8; ++i) {
      const int srow = w * 16 + i + 8 * hi;
      const int s = sb + srow;
      const int d = dt * 16 + lo;
      ctx_out[((size_t)s * BATCH + bb) * HID + hh * DHEAD + d] = cacc[dt][i];
    }
  }
}

// ---------------------------------------------------------------------------
extern "C" void kernel_launch(void* const* d_in, const int* in_sizes, int n_in,
                              void* d_out, int out_size, void* d_ws,
                              size_t ws_size, hipStream_t stream) {
  (void)in_sizes; (void)n_in; (void)out_size; (void)ws_size;

  const float* hidden = (const float*)d_in[0];  // [S, B, H]
  const float* W_qkv  = (const float*)d_in[1];  // [3H, H]
  const float* b_qkv  = (const float*)d_in[2];  // [3H]

  const size_t QSZ = (size_t)NM * S_LEN * DHEAD;  // 4,194,304 floats
  float* ws  = (float*)d_ws;
  float* Qb  = ws;
  float* Kb  = Qb + QSZ;
  float* Vb  = Kb + QSZ;
  float* sqv = Vb + QSZ;             // NM*S floats
  float* skv = sqv + (size_t)NM * S_LEN;

  float* ctx_out   = (float*)d_out;                          // S*B*H floats
  float* score_out = ctx_out + (size_t)S_LEN * BATCH * HID;  // NM*S*S floats

  // 1) QKV projection: C tiles 64(M) x 128(N), grid (3072/128, 4096/64)
  qkv_gemm_kernel<<<dim3(H3 / 128, (S_LEN * BATCH) / 64), 256, 0, stream>>>(
      hidden, W_qkv, b_qkv, Qb, Kb, Vb);

  // 2) row norms: 2 * NM * S rows
  row_norm_kernel<<<(2 * NM * S_LEN) / 256, 256, 0, stream>>>(Qb, Kb, sqv, skv);

  // 3) attention: grid (S/64 query blocks, NM slices), 128 threads
  attention_kernel<<<dim3(S_LEN / 64, NM), 128, 0, stream>>>(
      Qb, Kb, Vb, sqv, skv, ctx_out, score_out);
}